// DegreePrediction_28802050687271
// MI455X (gfx1250) — compile-verified
//
#include <hip/hip_runtime.h>
#include <math.h>

#define NN 80
#define PAIRS (NN * NN)
#define MAX_ITERS 60
#define TOLERANCE 1e-3f
#define SA 84                      // padded LDS row stride in floats (even -> 8B aligned b64 frags)
#define BLOCK_THREADS 160          // 5 waves of 32; wave w owns rows [16w, 16w+16)
#define KTILES 20                  // 80 / 4

typedef __attribute__((ext_vector_type(2))) float v2f;
typedef __attribute__((ext_vector_type(8))) float v8f;

// ---- one pass over w[0..79]: dot = v.w and sq = w.w; every wave computes identical totals
__device__ __forceinline__ void wave_dot_sq(const float* __restrict__ v,
                                            const float* __restrict__ w,
                                            int lane, float& dot, float& sq) {
    float d, s, wv;
    wv = w[lane];       d  = v[lane] * wv;       s  = wv * wv;
    wv = w[lane + 32];  d += v[lane + 32] * wv;  s += wv * wv;
    if (lane < 16) { wv = w[lane + 64]; d += v[lane + 64] * wv; s += wv * wv; }
    #pragma unroll
    for (int off = 16; off > 0; off >>= 1) {
        d += __shfl_xor(d, off, 32);
        s += __shfl_xor(s, off, 32);
    }
    dot = d; sq = s;
}

// ---- w = A * v via V_WMMA_F32_16X16X4_F32 with register-resident A fragments.
// v replicated into all 16 B columns, so every D column equals Av.
// Wave `wave` produces rows [16*wave, 16*wave+16).
__device__ __forceinline__ void matvec_wmma(const v2f* __restrict__ afrag,
                                            const float* __restrict__ vsrc,
                                            float* __restrict__ wdst,
                                            int wave, int lane, int khalf) {
    v8f acc = {};
    #pragma unroll
    for (int kt = 0; kt < KTILES; ++kt) {
        // B fragment 4x16 with every column = v: lane slot depends only on lane>>4
        v2f b = *(const v2f*)(vsrc + kt * 4 + khalf);
        acc = __builtin_amdgcn_wmma_f32_16x16x4_f32(false, afrag[kt], false, b,
                                                    (short)0, acc, false, false);
    }
    // Column 0 lives in lanes 0 (rows m=g) and 16 (rows m=8+g), VGPR g = acc[g]
    if ((lane & 15) == 0) {
        const int rbase = wave * 16 + (lane >> 4) * 8;
        #pragma unroll
        for (int g = 0; g < 8; ++g) wdst[rbase + g] = acc[g];
    }
}

__global__ void __launch_bounds__(BLOCK_THREADS)
rbc_power_iteration_kernel(const float* __restrict__ r_zeros,
                           const float* __restrict__ r_const,
                           const float* __restrict__ t_paths,
                           const float* __restrict__ weights_t,
                           const float* __restrict__ weights_r,
                           float* __restrict__ out) {
    __shared__ __align__(16) float Alds[NN * SA];
    __shared__ __align__(16) float vld[NN];
    __shared__ __align__(16) float wld[NN];

    const int p    = blockIdx.x;            // pair index (s,t); s = p / NN
    const int tid  = threadIdx.x;
    const int lane = tid & 31;
    const int wave = tid >> 5;

    // ---------- build A = weights_r * r_zeros + r_const into LDS (streamed once) ----------
    const size_t base = (size_t)p * (NN * NN);
    for (int i = tid * 4; i < NN * NN; i += BLOCK_THREADS * 4) {   // NN%4==0 -> no row crossing
        float4 wr = *(const float4*)(weights_r + base + i);
        float4 rz = *(const float4*)(r_zeros  + base + i);
        float4 rc = *(const float4*)(r_const  + base + i);
        float4 a;
        a.x = fmaf(wr.x, rz.x, rc.x);
        a.y = fmaf(wr.y, rz.y, rc.y);
        a.z = fmaf(wr.z, rz.z, rc.z);
        a.w = fmaf(wr.w, rz.w, rc.w);
        const int r = i / NN, c = i - r * NN;
        *(float4*)(Alds + r * SA + c) = a;                         // 16B aligned (SA%4==0, c%4==0)
    }
    if (tid < NN) vld[tid] = 0.11180339887498949f;                 // 1/sqrt(80)
    __syncthreads();

    // ---------- hoist this lane's A fragments into VGPRs (invariant across iterations) ----
    // A fragment layout for 16x4 f32: lane holds A[row][k0], A[row][k0+1],
    // row = 16*wave + lane%16, k0 = 4*kt + 2*(lane/16).
    const int row   = wave * 16 + (lane & 15);
    const int khalf = (lane >> 4) * 2;
    v2f afrag[KTILES];
    {
        const float* arow = Alds + row * SA + khalf;
        #pragma unroll
        for (int kt = 0; kt < KTILES; ++kt)
            afrag[kt] = *(const v2f*)(arow + kt * 4);
    }

    // ---------- power iteration: carry w = A*v so each step costs one WMMA matvec ----------
    matvec_wmma(afrag, vld, wld, wave, lane, khalf);               // w0 = A v0
    __syncthreads();
    float ev, nrm2;
    wave_dot_sq(vld, wld, lane, ev, nrm2);                         // ev0 = v0.w0, ||w0||^2

    for (int it = 0; it < MAX_ITERS; ++it) {
        const float inv = 1.0f / sqrtf(nrm2);
        __syncthreads();                                           // done reading vld/wld above
        if (tid < NN) vld[tid] = wld[tid] * inv;                   // v = w / ||w||
        __syncthreads();
        matvec_wmma(afrag, vld, wld, wave, lane, khalf);           // w = A v
        __syncthreads();
        float ev_new, n2;
        wave_dot_sq(vld, wld, lane, ev_new, n2);                   // ev = v.Av, ||w||^2 for next
        const bool conv = fabsf(ev - ev_new) < TOLERANCE;          // uniform across block
        ev = ev_new; nrm2 = n2;
        if (conv) break;                                           // reference freezes AFTER update
    }

    // ---------- deltas: v * (T / v[src]) accumulated into global rbc ----------
    const float Tval  = weights_t[p] * t_paths[p];
    const float scale = Tval / vld[p / NN];
    if (tid < NN) atomicAdd(out + tid, vld[tid] * scale);
}

__global__ void zero_out_kernel(float* __restrict__ out) {
    if (threadIdx.x < NN) out[threadIdx.x] = 0.0f;
}

extern "C" void kernel_launch(void* const* d_in, const int* in_sizes, int n_in,
                              void* d_out, int out_size, void* d_ws, size_t ws_size,
                              hipStream_t stream) {
    // inputs: 0=x (unused by reference), 1=r_zeros, 2=r_const, 3=t_paths, 4=weights_t, 5=weights_r
    const float* r_zeros   = (const float*)d_in[1];
    const float* r_const   = (const float*)d_in[2];
    const float* t_paths   = (const float*)d_in[3];
    const float* weights_t = (const float*)d_in[4];
    const float* weights_r = (const float*)d_in[5];
    float* out = (float*)d_out;

    zero_out_kernel<<<1, 128, 0, stream>>>(out);
    rbc_power_iteration_kernel<<<PAIRS, BLOCK_THREADS, 0, stream>>>(
        r_zeros, r_const, t_paths, weights_t, weights_r, out);
}